// MCR2_55233279426744
// MI455X (gfx1250) — compile-verified
//
#include <hip/hip_runtime.h>
#include <math.h>

// ---------------------------------------------------------------------------
// MCR^2 loss on MI455X (gfx1250): 11x [32x32] Grams from Z[600000x32] f32,
// per-class masked accumulation via V_WMMA_F32_16X16X4_F32, then wave-parallel
// log-dets.  Memory-bound problem (~77MB @ 23.3TB/s ~ 3.4us floor).
// ---------------------------------------------------------------------------

typedef float v2f __attribute__((ext_vector_type(2)));
typedef float v8f __attribute__((ext_vector_type(8)));

#define NCLS 10
#define DIM  32
#define MAX_WG 512

// ---------------------------------------------------------------------------
// Phase 1: per-workgroup partial Grams.
// Block = 128 threads = 4 waves; wave w owns Gram tile (di=w>>1, dj=w&1).
// WMMA f32 16x16x4: K = sample dimension, 4 samples per chunk.
//   A (16x4, MxK) layout: lane m=lane&15; lanes<16 -> {K0,K1} in {v0,v1},
//                         lanes>=16 -> {K2,K3}.   B (4x16) mirrors it.
// ---------------------------------------------------------------------------
__global__ void __launch_bounds__(128) mcr2_gram_kernel(
    const float* __restrict__ Z, const int* __restrict__ labels,
    int n, int nchunks,
    float* __restrict__ partial,   // [num_wg][NCLS][DIM*DIM]
    int*   __restrict__ pcnt)      // [num_wg][16]  (NCLS used)
{
  const int tid  = threadIdx.x;
  const int lane = tid & 31;
  const int wave = tid >> 5;
  const int di   = wave >> 1;
  const int dj   = wave & 1;
  const int m    = lane & 15;   // M (for A) / N (for B) index within tile
  const int kh   = lane >> 4;   // 0 -> samples {0,1}, 1 -> samples {2,3}
  const int colA = di * 16 + m;
  const int colB = dj * 16 + m;

  v8f acc[NCLS];
#pragma unroll
  for (int j = 0; j < NCLS; ++j)
#pragma unroll
    for (int r = 0; r < 8; ++r) acc[j][r] = 0.0f;

  int cnt[NCLS];
#pragma unroll
  for (int j = 0; j < NCLS; ++j) cnt[j] = 0;

  for (int c = blockIdx.x; c < nchunks; c += gridDim.x) {
    const int s = c * 4 + 2 * kh;          // sample row for lane's K-pair

    // prefetch next chunk's rows (speculative, CDNA5 global_prefetch_b8)
    if (c + (int)gridDim.x < nchunks)
      __builtin_prefetch(Z + (size_t)(c + gridDim.x) * 4 * DIM + lane * 4, 0, 1);

    float ax = 0.0f, ay = 0.0f, bx = 0.0f, by = 0.0f;
    int la = -1, lb = -1;
    if (s < n) {
      ax = Z[(size_t)s * DIM + colA];
      bx = Z[(size_t)s * DIM + colB];
      la = labels[s];
    }
    if (s + 1 < n) {
      ay = Z[(size_t)(s + 1) * DIM + colA];
      by = Z[(size_t)(s + 1) * DIM + colB];
      lb = labels[s + 1];
    }

    // wave-uniform bitmask of classes present among the 4 samples:
    // lane 0 sees labels(s0,s1), lane 16 sees labels(s2,s3).
    unsigned pm = 0;
    if (la >= 0) pm |= 1u << la;
    if (lb >= 0) pm |= 1u << lb;
    const unsigned comb =
        (unsigned)__builtin_amdgcn_readlane((int)pm, 0) |
        (unsigned)__builtin_amdgcn_readlane((int)pm, 16);

    v2f a; a.x = ax; a.y = ay;

#pragma unroll
    for (int j = 0; j < NCLS; ++j) {
      if (comb & (1u << j)) {              // scalar branch: EXEC stays full
        v2f bm;
        bm.x = (la == j) ? bx : 0.0f;
        bm.y = (lb == j) ? by : 0.0f;
        acc[j] = __builtin_amdgcn_wmma_f32_16x16x4_f32(
            false, a, false, bm, (short)0, acc[j], false, false);
      }
    }

    if (wave == 0 && m == 0) {             // lanes 0 & 16 together see all 4 labels
#pragma unroll
      for (int j = 0; j < NCLS; ++j)
        cnt[j] += ((la == j) ? 1 : 0) + ((lb == j) ? 1 : 0);
    }
  }

  // C/D layout: lane n=lane&15, VGPR r holds C[(lane>>4)*8 + r][n]
  float* p = partial + (size_t)blockIdx.x * (NCLS * DIM * DIM);
  const int rowg = di * 16 + kh * 8;
  const int colg = dj * 16 + m;
#pragma unroll
  for (int j = 0; j < NCLS; ++j)
#pragma unroll
    for (int r = 0; r < 8; ++r)
      p[j * (DIM * DIM) + (rowg + r) * DIM + colg] = acc[j][r];

  if (wave == 0) {
#pragma unroll
    for (int j = 0; j < NCLS; ++j) {
      int tot = __builtin_amdgcn_readlane(cnt[j], 0) +
                __builtin_amdgcn_readlane(cnt[j], 16);
      if (tid == 0) pcnt[(size_t)blockIdx.x * 16 + j] = tot;
    }
  }
}

// ---------------------------------------------------------------------------
// Uniform lane-broadcast via v_readlane_b32 (avoids ds_bpermute that __shfl
// emits; the source lane is a compile-time constant after unrolling).
// ---------------------------------------------------------------------------
__device__ __forceinline__ float readlane_f32(float v, int srclane) {
  return __int_as_float(__builtin_amdgcn_readlane(__float_as_int(v), srclane));
}

// ---------------------------------------------------------------------------
// Wave-parallel logdet of a 32x32 matrix: lane j holds column j in a[0..31].
// Gaussian elimination without pivoting (matrix is I + c*PSD, diag-dominant).
// ---------------------------------------------------------------------------
__device__ __forceinline__ float lu_logdet32(float* a) {
  float ld = 0.0f;
#pragma unroll
  for (int k = 0; k < DIM; ++k) {
    const float pk = readlane_f32(a[k], k);  // pivot A[k][k] lives in lane k
    ld += __logf(pk);
    const float rpk = 1.0f / pk;
#pragma unroll
    for (int i = k + 1; i < DIM; ++i) {
      const float f = readlane_f32(a[i], k) * rpk;  // A[i][k] / A[k][k]
      a[i] = fmaf(-f, a[k], a[i]);
    }
  }
  return ld;
}

// ---------------------------------------------------------------------------
// Phase 2: block j reduces class-j partials (fixed order -> deterministic),
// writes the final class Gram, and computes logdet(I + (D/(trPi_j*eps))*Gj).
// D/eps = 32/0.5 = 64.
// ---------------------------------------------------------------------------
__global__ void __launch_bounds__(32) mcr2_reduce_logdet_kernel(
    const float* __restrict__ partial, const int* __restrict__ pcnt,
    int num_wg,
    float* __restrict__ gram_final,    // [NCLS][DIM*DIM]
    float* __restrict__ stats)         // [0..9]=logdet_j  [10..19]=trPi_j
{
  const int cls  = blockIdx.x;
  const int lane = threadIdx.x;

  int tot = 0;
  for (int w = 0; w < num_wg; ++w) tot += pcnt[(size_t)w * 16 + cls];

  float s[DIM];
#pragma unroll
  for (int i = 0; i < DIM; ++i) s[i] = 0.0f;
  for (int w = 0; w < num_wg; ++w) {
    const float* p = partial + (size_t)w * (NCLS * DIM * DIM) + cls * (DIM * DIM);
#pragma unroll
    for (int i = 0; i < DIM; ++i) s[i] += p[i * DIM + lane];
  }
#pragma unroll
  for (int i = 0; i < DIM; ++i)
    gram_final[cls * (DIM * DIM) + i * DIM + lane] = s[i];

  const float trpi = (float)tot;
  const float csc  = (tot > 0) ? (64.0f / trpi) : 0.0f;
  float a[DIM];
#pragma unroll
  for (int i = 0; i < DIM; ++i)
    a[i] = ((i == lane) ? 1.0f : 0.0f) + csc * s[i];

  const float ld = lu_logdet32(a);
  if (lane == 0) {
    stats[cls]        = ld;
    stats[NCLS + cls] = trpi;
  }
}

// ---------------------------------------------------------------------------
// Phase 3: G = sum_j Gj, logdet(I + (64/n) G), combine into 3 outputs.
// ---------------------------------------------------------------------------
__global__ void __launch_bounds__(32) mcr2_final_kernel(
    const float* __restrict__ gram_final, const float* __restrict__ stats,
    int n, float* __restrict__ out)
{
  const int lane = threadIdx.x;
  float s[DIM];
#pragma unroll
  for (int i = 0; i < DIM; ++i) s[i] = 0.0f;
#pragma unroll
  for (int j = 0; j < NCLS; ++j)
#pragma unroll
    for (int i = 0; i < DIM; ++i)
      s[i] += gram_final[j * (DIM * DIM) + i * DIM + lane];

  const float c = 64.0f / (float)n;
  float a[DIM];
#pragma unroll
  for (int i = 0; i < DIM; ++i)
    a[i] = ((i == lane) ? 1.0f : 0.0f) + c * s[i];

  const float loss_R = 0.5f * lu_logdet32(a);

  float loss_Rc = 0.0f;
  const float inv2n = 1.0f / (2.0f * (float)n);
#pragma unroll
  for (int j = 0; j < NCLS; ++j)
    loss_Rc += stats[NCLS + j] * inv2n * stats[j];

  if (lane == 0) {
    out[0] = -(loss_R - loss_Rc);
    out[1] = loss_R;
    out[2] = loss_Rc;
  }
}

// ---------------------------------------------------------------------------
extern "C" void kernel_launch(void* const* d_in, const int* in_sizes, int n_in,
                              void* d_out, int out_size, void* d_ws, size_t ws_size,
                              hipStream_t stream) {
  const float* Z      = (const float*)d_in[0];
  const int*   labels = (const int*)d_in[1];
  const int n       = in_sizes[1];        // number of samples (rows of Z)
  const int nchunks = (n + 3) / 4;

  // ws layout: [gram_final 10*1024 f][stats 32 f][partials ...][counts ...]
  char* ws = (char*)d_ws;
  float* gram_final = (float*)ws;
  float* stats      = (float*)(ws + (size_t)NCLS * DIM * DIM * 4);
  char*  dyn        = ws + (size_t)NCLS * DIM * DIM * 4 + 128;

  const size_t fixed  = (size_t)NCLS * DIM * DIM * 4 + 128;
  const size_t per_wg = (size_t)NCLS * DIM * DIM * 4 + 16 * 4;
  size_t avail = (ws_size > fixed) ? (ws_size - fixed) : 0;
  int num_wg = (int)(avail / per_wg);
  if (num_wg < 1) num_wg = 1;
  if (num_wg > MAX_WG) num_wg = MAX_WG;
  if (num_wg > nchunks) num_wg = nchunks;

  float* partial = (float*)dyn;
  int*   pcnt    = (int*)(dyn + (size_t)num_wg * NCLS * DIM * DIM * 4);

  mcr2_gram_kernel<<<num_wg, 128, 0, stream>>>(Z, labels, n, nchunks,
                                               partial, pcnt);
  mcr2_reduce_logdet_kernel<<<NCLS, 32, 0, stream>>>(partial, pcnt, num_wg,
                                                     gram_final, stats);
  mcr2_final_kernel<<<1, 32, 0, stream>>>(gram_final, stats, n,
                                          (float*)d_out);
}